// SphericalHarmonics_1417339208114
// MI455X (gfx1250) — compile-verified
//
#include <hip/hip_runtime.h>
#include <stdint.h>

// ---------------------------------------------------------------------------
// Compile-time tables: K(l,m) normalization and Legendre recurrence coeffs.
// ---------------------------------------------------------------------------
namespace sh {

constexpr int L = 24;
constexpr double PI_D = 3.14159265358979323846264338327950288;

constexpr double csqrt(double x) {
  if (x <= 0.0) return 0.0;
  double s = 1.0, v = x;
  while (v < 0.25) { v *= 4.0; s *= 0.5; }
  while (v > 4.0)  { v *= 0.25; s *= 2.0; }
  double g = v;
  for (int i = 0; i < 40; ++i) g = 0.5 * (g + v / g);
  return g * s;
}

struct Tab {
  float c0[L];      // K(l,0)
  float cp[L][L];   // sqrt(2)*K(l,m), m>=1
  float A[L][L];    // (2l-1)/(l-m)     for m<=l-2
  float B[L][L];    // (l+m-1)/(l-m)    for m<=l-2
};

constexpr Tab make() {
  Tab t{};
  for (int l = 0; l < L; ++l) {
    for (int m = 0; m <= l; ++m) {
      double invprod = 1.0;
      for (int i = l - m + 1; i <= l + m; ++i) invprod /= (double)i;
      double K = csqrt((2.0 * l + 1.0) * invprod / (4.0 * PI_D));
      if (m == 0) t.c0[l] = (float)K;
      else        t.cp[l][m] = (float)(csqrt(2.0) * K);
    }
    for (int m = 0; m <= l - 2; ++m) {
      t.A[l][m] = (float)((2.0 * (double)l - 1.0) / (double)(l - m));
      t.B[l][m] = (float)(((double)l + (double)m - 1.0) / (double)(l - m));
    }
  }
  return t;
}

__device__ constexpr Tab TAB = make();

} // namespace sh

#define TPB   128           // 4 waves; 3125 blocks * 128 = 400000 exactly
#define MAXW  47            // 2*23+1

#if defined(__HIP_DEVICE_COMPILE__) && defined(__gfx1250__) && \
    __has_builtin(__builtin_amdgcn_tensor_store_from_lds) && \
    __has_builtin(__builtin_amdgcn_s_wait_tensorcnt)
#define USE_TDM 1
#else
#define USE_TDM 0
#endif

typedef unsigned int v4u __attribute__((ext_vector_type(4)));
typedef int          v8i __attribute__((ext_vector_type(8)));
typedef int          v4i __attribute__((ext_vector_type(4)));

__global__ __launch_bounds__(TPB)
void sh_kernel(const float* __restrict__ rd, float* __restrict__ out, int N) {
  __shared__ float sbuf[2][TPB * MAXW];   // 48,128 B double buffer

  const int tid     = threadIdx.x;
  const int base_pt = blockIdx.x * TPB;
  int rows = N - base_pt; if (rows > TPB) rows = TPB;
  int n = base_pt + tid;  if (n >= N) n = N - 1;

  const float d2r = 0.017453292519943295f;
  const float ra  = rd[2 * n + 0];
  const float dec = rd[2 * n + 1];
  const float phi   = ra * d2r;
  const float theta = (dec + 90.0f) * d2r;
  const float x     = cosf(theta);
  const float somx2 = sqrtf((1.0f - x) * (1.0f + x));
  const float c1 = cosf(phi), s1 = sinf(phi);

  // cos(m*phi), sin(m*phi) via angle-addition recurrence (registers)
  float cm[sh::L], smv[sh::L];
  cm[0] = 1.0f; smv[0] = 0.0f;
  cm[1] = c1;   smv[1] = s1;
#pragma unroll
  for (int m = 2; m < sh::L; ++m) {
    cm[m]  = cm[m - 1] * c1 - smv[m - 1] * s1;
    smv[m] = smv[m - 1] * c1 + cm[m - 1] * s1;
  }

  // l-outer Legendre: P1[m] = P(l-1,m), P2[m] = P(l-2,m)
  float P1[sh::L] = {};
  float P2[sh::L] = {};

#pragma unroll
  for (int l = 0; l < sh::L; ++l) {
    const int W = 2 * l + 1;
    float* buf = sbuf[l & 1];

#if USE_TDM
    if (l >= 2) {
      // buffer (l&1) was consumed by the tensor store issued at step l-2;
      // allow only the (l-1) store to remain outstanding before reuse.
      if (tid < 32) __builtin_amdgcn_s_wait_tensorcnt(1);
      __syncthreads();
    }
#endif

    float Pl[sh::L];
#pragma unroll
    for (int m = 0; m <= l - 2; ++m)
      Pl[m] = sh::TAB.A[l][m] * x * P1[m] - sh::TAB.B[l][m] * P2[m];
    if (l >= 1) Pl[l - 1] = x * (float)(2 * l - 1) * P1[l - 1];
    Pl[l] = (l == 0) ? 1.0f : P1[l - 1] * (-(float)(2 * l - 1)) * somx2;

    // write this l-row into LDS, packed [point][2l+1]; odd stride => no bank conflicts
    float* row = buf + tid * W + l;   // center (m = 0) at j = l
    row[0] = sh::TAB.c0[l] * Pl[0];
#pragma unroll
    for (int m = 1; m <= l; ++m) {
      const float t1 = sh::TAB.cp[l][m] * Pl[m];
      row[ m] = t1 * cm[m];    // y(l, +m)
      row[-m] = t1 * smv[m];   // y(l, -m)
    }

#pragma unroll
    for (int m = 0; m <= l; ++m) { P2[m] = P1[m]; P1[m] = Pl[m]; }

    __syncthreads();  // LDS slab complete

#if USE_TDM
    // wave 0 issues one async TDM 2D tile store: [rows x W] tile, row stride 576
    if (tid < 32) {
      const uint32_t lds_base = (uint32_t)(uintptr_t)&buf[0];
      const uint64_t ga = (uint64_t)(uintptr_t)
          (out + (size_t)base_pt * 576u + (size_t)(l * l));
      const uint32_t W32 = (uint32_t)W;
      const uint32_t R32 = (uint32_t)rows;

      // D# group 0: count=1 | lds_addr | global_addr[56:0] | type=2
      v4u g0 = { 1u,
                 lds_base,
                 (uint32_t)(ga & 0xFFFFFFFFu),
                 (uint32_t)((ga >> 32) & 0x1FFFFFFull) | (2u << 30) };

      // D# group 1: data_size=4B; tensor_dim0=W, tensor_dim1=rows;
      // tile_dim0=W, tile_dim1=rows, tile_dim2=0; dim0_stride=576; dim1_stride=0
      v8i g1 = { (int)(2u << 16),                                      // d0
                 (int)((W32 & 0xFFFFu) << 16),                         // d1
                 (int)(((W32 >> 16) & 0xFFFFu) | ((R32 & 0xFFFFu) << 16)), // d2
                 (int)(((R32 >> 16) & 0xFFFFu) | ((W32 & 0xFFFFu) << 16)), // d3
                 (int)(R32 & 0xFFFFu),                                 // d4
                 (int)576,                                             // d5
                 0, 0 };                                               // d6, d7
      v4i gz4 = { 0, 0, 0, 0 };
      v8i gz8 = { 0, 0, 0, 0, 0, 0, 0, 0 };
      __builtin_amdgcn_tensor_store_from_lds(g0, g1, gz4, gz4, gz8, 0);
    }
#else
    // Fallback: cooperative coalesced copy (consecutive lanes -> consecutive addrs)
    const int total = rows * W;
    for (int k = tid; k < total; k += TPB) {
      const int r = k / W;
      const int j = k - r * W;
      out[(size_t)(base_pt + r) * 576u + (size_t)(l * l) + j] = buf[k];
    }
    __syncthreads();
#endif
  }
  // S_ENDPGM performs an implicit wait-idle (drains TENSORcnt).
}

extern "C" void kernel_launch(void* const* d_in, const int* in_sizes, int n_in,
                              void* d_out, int out_size, void* d_ws, size_t ws_size,
                              hipStream_t stream) {
  (void)n_in; (void)out_size; (void)d_ws; (void)ws_size;
  const float* rd  = (const float*)d_in[0];
  float*       out = (float*)d_out;
  const int N = in_sizes[0] / 2;              // [N,2] ra/dec pairs
  const int blocks = (N + TPB - 1) / TPB;     // 3125 for N=400000
  sh_kernel<<<blocks, TPB, 0, stream>>>(rd, out, N);
}